// FFTLongConv_28269474742360
// MI455X (gfx1250) — compile-verified
//
#include <hip/hip_runtime.h>

// FFT long conv for MI455X (gfx1250):
//   B=4, H=8, D=64, L=8192, N=2L=16384 = 16*16*16*4
//
// One workgroup per (h,d) pair (512 blocks). Each WG:
//   1) F = FFT(filter row) once (filters are shared across B) -> kept in LDS.
//   2) For each b-pair (0,1),(2,3): pack z = x_b0 + i*x_b1 (real-filter
//      linearity => IFFT(FFT(z)*F) = y_b0 + i*y_b1), forward FFT, pointwise
//      multiply in the digit-reversed domain, inverse FFT, masked epilogue.
// => 5 FFTs per 4 rows instead of 3 FFTs per row.
//
// Radix-16 stages run on the f32 matrix pipe (V_WMMA_F32_16X16X4_F32).
// Forward = DIF (natural -> digit-reversed), inverse = DIT with conjugate
// twiddles; the pointwise product happens in the scrambled domain so no
// bit-reversal pass exists. Complex data is interleaved (v2f) so every LDS
// element access is one ds_load_b64/ds_store_b64; twiddles come from a
// quarter-wave (cos,sin) table (one b64 load + quadrant fixup).
// DFT16 coefficients come from an exact 16-entry constant table; the twiddle
// table is built as miniA x miniB unit-complex products (128 sincos/block).
// Batch loop unrolls by 2 only: ~120 live VGPRs, no scratch spills, while
// keeping four independent WMMA accumulation chains in flight per wave.
// LDS: 128KB (Z) + 128KB (F) + 32KB (table) + 1KB (mini) = 289KB.

#define N_FFT   16384
#define N_HALF  8192
#define NMASK   16383
#define QMASK   4095

typedef float v2f __attribute__((ext_vector_type(2)));
typedef float v8f __attribute__((ext_vector_type(8)));

// cos(2*pi*k/16), k = 0..15 (exact DFT16 coefficient set).
__device__ const float DFT16C[16] = {
     1.0f,                    0.92387953251128675613f,
     0.70710678118654752440f, 0.38268343236508977173f,
     0.0f,                   -0.38268343236508977173f,
    -0.70710678118654752440f,-0.92387953251128675613f,
    -1.0f,                   -0.92387953251128675613f,
    -0.70710678118654752440f,-0.38268343236508977173f,
     0.0f,                    0.38268343236508977173f,
     0.70710678118654752440f, 0.92387953251128675613f};

__device__ __forceinline__ v8f wmma_f32x4(v2f a, v2f b, v8f c) {
#if defined(__HIP_DEVICE_COMPILE__)
  // (neg_a, A, neg_b, B, c_mod, C, reuse_a, reuse_b) -> v_wmma_f32_16x16x4_f32
  return __builtin_amdgcn_wmma_f32_16x16x4_f32(false, a, false, b, (short)0, c,
                                               false, false);
#else
  (void)a; (void)b;
  return c;
#endif
}

// (cos, sin)(2*pi*k/N) from a quarter-period (cos,sin) pair table.
__device__ __forceinline__ void tw_cs(const v2f* tab, int k, float& c,
                                      float& s) {
  const int kk = k & NMASK;
  const v2f t = tab[kk & QMASK];
  const bool sw = (kk & 4096) != 0;   // rotate by pi/2: (c,s) -> (-s,c)
  float cc = sw ? -t.y : t.x;
  float ss = sw ? t.x : t.y;
  if (kk & 8192) { cc = -cc; ss = -ss; }  // rotate by pi
  c = cc;
  s = ss;
}

// (r + i*x) *= exp(-2*pi*i*k/N). Negative k gives the conjugate twiddle.
__device__ __forceinline__ void crot(float& r, float& x, int k,
                                     const v2f* tab) {
  float c, s;
  tw_cs(tab, k, c, s);
  const float nr = r * c + x * s;
  const float nx = x * c - r * s;
  r = nr;
  x = nx;
}

// One radix-16 stage over a 16K complex array.
//  dif=true : DIF butterfly (DFT16 then twiddle W_S^{i*j}), forward sign.
//  dif=false: DIT butterfly (twiddle W_S^{+i*m} then conj-DFT16), inverse.
// 1024 butterflies/stage = 64 batches of 16; 16 waves * 4 batches each.
// Batch = complex 16x16x16 matmul = 16 chained v_wmma_f32_16x16x4_f32.
__device__ __forceinline__ void radix16_stage(
    v2f* Z, const v2f* tab, int ls /* log2(s) */, bool fwd, bool dif,
    const v2f Ac[4], const v2f As[4], const v2f AsN[4], int wave, int lane)
{
  const int s      = 1 << ls;
  const int twstep = 1024 >> ls;          // N / S, S = 16*s
  const int g      = lane & 15;
  const int hi     = lane >> 4;

  const v2f* S1 = fwd ? As : AsN;  // Dr = Ac*Xr + S1*Xi
  const v2f* S2 = fwd ? AsN : As;  // Di = Ac*Xi + S2*Xr

  #pragma unroll 2
  for (int q = 0; q < 4; ++q) {
    const int t    = ((wave + (q << 4)) << 4) + g;   // butterfly id 0..1023
    const int blk  = t >> ls;
    const int i    = t & (s - 1);
    const int base = (blk << (ls + 4)) + i;          // blk*S + i

    // Load B operand chunks (K-chunks of 4): lane holds (m0,g) and (m0+1,g).
    v2f Br[4], Bi[4];
    #pragma unroll
    for (int c = 0; c < 4; ++c) {
      const int m0 = 4 * c + (hi ? 2 : 0);
      const int i0 = base + (m0 << ls);
      const v2f e0 = Z[i0];
      const v2f e1 = Z[i0 + s];
      float r0 = e0.x, x0 = e0.y;
      float r1 = e1.x, x1 = e1.y;
      if (!dif) { // DIT inverse: pre-twiddle by exp(+2pi*i * i*m*twstep / N)
        crot(r0, x0, -(i * m0 * twstep), tab);
        crot(r1, x1, -(i * (m0 + 1) * twstep), tab);
      }
      v2f tr; tr.x = r0; tr.y = r1; Br[c] = tr;
      v2f ti; ti.x = x0; ti.y = x1; Bi[c] = ti;
    }

    v8f Dr = {};
    v8f Di = {};
    #pragma unroll
    for (int c = 0; c < 4; ++c) Dr = wmma_f32x4(Ac[c], Br[c], Dr);
    #pragma unroll
    for (int c = 0; c < 4; ++c) Dr = wmma_f32x4(S1[c], Bi[c], Dr);
    #pragma unroll
    for (int c = 0; c < 4; ++c) Di = wmma_f32x4(Ac[c], Bi[c], Di);
    #pragma unroll
    for (int c = 0; c < 4; ++c) Di = wmma_f32x4(S2[c], Br[c], Di);

    // D layout: VGPR v -> row j = v (lanes 0-15) / v+8 (lanes 16-31).
    #pragma unroll
    for (int v = 0; v < 8; ++v) {
      const int j = v + (hi ? 8 : 0);
      float yr = Dr[v], yi = Di[v];
      if (dif) { // DIF forward: post-twiddle by exp(-2pi*i * i*j*twstep / N)
        crot(yr, yi, i * j * twstep, tab);
      }
      v2f o; o.x = yr; o.y = yi;
      Z[base + (j << ls)] = o;
    }
  }
}

// Final forward / first inverse radix-4 stage (trivial twiddles) on VALU.
__device__ __forceinline__ void radix4_stage(v2f* Z, bool fwd, int tid) {
  #pragma unroll 2
  for (int q = 0; q < 8; ++q) {
    const int p = (tid + (q << 9)) << 2;
    const v2f a0 = Z[p], a1 = Z[p + 1], a2 = Z[p + 2], a3 = Z[p + 3];
    const float e0r = a0.x + a2.x, e0i = a0.y + a2.y;
    const float e1r = a0.x - a2.x, e1i = a0.y - a2.y;
    const float o0r = a1.x + a3.x, o0i = a1.y + a3.y;
    const float o1r = a1.x - a3.x, o1i = a1.y - a3.y;
    v2f y0; y0.x = e0r + o0r; y0.y = e0i + o0i;
    v2f y2; y2.x = e0r - o0r; y2.y = e0i - o0i;
    v2f y1, y3;
    if (fwd) { // W4 = -i
      y1.x = e1r + o1i; y1.y = e1i - o1r;
      y3.x = e1r - o1i; y3.y = e1i + o1r;
    } else {   // conj
      y1.x = e1r - o1i; y1.y = e1i + o1r;
      y3.x = e1r + o1i; y3.y = e1i - o1r;
    }
    Z[p] = y0; Z[p + 1] = y1; Z[p + 2] = y2; Z[p + 3] = y3;
  }
}

__global__ __launch_bounds__(512, 1)
void fft_long_conv_kernel(const float* __restrict__ inputs,
                          const float* __restrict__ filters,
                          const float* __restrict__ bias,
                          const long long* __restrict__ positions,
                          float* __restrict__ out)
{
  __shared__ v2f sZ[N_FFT];      // packed pair of input rows / work array
  __shared__ v2f sF[N_FFT];      // filter spectrum (lives all block long)
  __shared__ v2f sTab[4096];     // (cos,sin)(2*pi*r/N), r in [0, N/4)
  __shared__ v2f sMini[128];     // [0..63]: coarse e^{i*2pi*64a/N}; [64..127]: fine

  const int tid    = threadIdx.x;
  const int row512 = blockIdx.x;          // h*64 + d
  const int d      = row512 & 63;

  const float* fg = filters + (size_t)row512 * N_HALF;

  for (int l = tid; l < N_HALF; l += 512) {
    v2f v; v.x = fg[l]; v.y = 0.f;
    sF[l] = v;
    v2f z; z.x = 0.f; z.y = 0.f;
    sF[l + N_HALF] = z;
  }
  // Mini twiddle tables: only 128 sincos evaluations per block.
  if (tid < 64) {
    const float thA = 6.28318530717958647692f * (float)(tid << 6) / (float)N_FFT;
    const float thB = 6.28318530717958647692f * (float)tid / (float)N_FFT;
    v2f a; a.x = cosf(thA); a.y = sinf(thA); sMini[tid]      = a;
    v2f b; b.x = cosf(thB); b.y = sinf(thB); sMini[tid + 64] = b;
  }
  __syncthreads();
  // Quarter-wave table as coarse x fine unit-complex products.
  for (int r = tid; r < 4096; r += 512) {
    const v2f a = sMini[r >> 6];
    const v2f b = sMini[(r & 63) + 64];
    v2f t;
    t.x = a.x * b.x - a.y * b.y;
    t.y = a.x * b.y + a.y * b.x;
    sTab[r] = t;
  }

  // DFT16 coefficient matrices in the WMMA A-operand layout, from the exact
  // 16-entry table: lane -> row j = lane&15; VGPRs cover K = 4c + {hi?2:0,+1}.
  const int lane = tid & 31;
  const int wave = tid >> 5;
  const int jrow = lane & 15;
  const int hi   = lane >> 4;
  v2f Ac[4], As[4], AsN[4];
  #pragma unroll
  for (int c = 0; c < 4; ++c) {
    const int m0 = 4 * c + (hi ? 2 : 0);
    const int k0 = (jrow * m0) & 15;
    const int k1 = (jrow * (m0 + 1)) & 15;
    v2f vc; vc.x = DFT16C[k0];            vc.y = DFT16C[k1];            Ac[c]  = vc;
    v2f vs; vs.x = DFT16C[(k0 + 12) & 15]; vs.y = DFT16C[(k1 + 12) & 15]; As[c] = vs;
    v2f vn; vn.x = -vs.x;                 vn.y = -vs.y;                 AsN[c] = vn;
  }
  __syncthreads();

  // Filter spectrum, computed once per (h,d): DIF s = 1024, 64, 4, radix-4.
  radix16_stage(sF, sTab, 10, true, true, Ac, As, AsN, wave, lane); __syncthreads();
  radix16_stage(sF, sTab, 6,  true, true, Ac, As, AsN, wave, lane); __syncthreads();
  radix16_stage(sF, sTab, 2,  true, true, Ac, As, AsN, wave, lane); __syncthreads();
  radix4_stage(sF, true, tid); __syncthreads();

  const float sc = 1.0f / ((float)N_FFT * (float)N_FFT);
  const float bv = bias[d];

  #pragma unroll 1
  for (int p = 0; p < 2; ++p) {
    const int b0 = 2 * p, b1 = 2 * p + 1;
    const float* x0 = inputs + ((size_t)b0 * 512 + row512) * N_HALF;
    const float* x1 = inputs + ((size_t)b1 * 512 + row512) * N_HALF;

    // z = x_b0 + i * x_b1, zero-padded to 2L.
    for (int l = tid; l < N_HALF; l += 512) {
      v2f v; v.x = x0[l]; v.y = x1[l];
      sZ[l] = v;
      v2f z; z.x = 0.f; z.y = 0.f;
      sZ[l + N_HALF] = z;
    }
    __syncthreads();

    // Forward FFT of z (DIF, natural -> digit-reversed).
    radix16_stage(sZ, sTab, 10, true, true, Ac, As, AsN, wave, lane); __syncthreads();
    radix16_stage(sZ, sTab, 6,  true, true, Ac, As, AsN, wave, lane); __syncthreads();
    radix16_stage(sZ, sTab, 2,  true, true, Ac, As, AsN, wave, lane); __syncthreads();
    radix4_stage(sZ, true, tid); __syncthreads();

    // Pointwise product in the shared digit-reversed domain, with 1/N^2
    // (reference scaling: y = IDFT_unscaled(Z*F) / N^2).
    for (int l = tid; l < N_FFT; l += 512) {
      const v2f z = sZ[l], f = sF[l];
      v2f o;
      o.x = (z.x * f.x - z.y * f.y) * sc;
      o.y = (z.x * f.y + z.y * f.x) * sc;
      sZ[l] = o;
    }
    __syncthreads();

    // Unscaled inverse FFT (DIT mirror: radix-4, s = 4, 64, 1024).
    radix4_stage(sZ, false, tid); __syncthreads();
    radix16_stage(sZ, sTab, 2,  false, false, Ac, As, AsN, wave, lane); __syncthreads();
    radix16_stage(sZ, sTab, 6,  false, false, Ac, As, AsN, wave, lane); __syncthreads();
    radix16_stage(sZ, sTab, 10, false, false, Ac, As, AsN, wave, lane); __syncthreads();

    // Epilogue: real part -> row b0, imag part -> row b1.
    float* o0 = out + ((size_t)b0 * 512 + row512) * N_HALF;
    float* o1 = out + ((size_t)b1 * 512 + row512) * N_HALF;
    const long long* p0 = positions + (size_t)b0 * N_HALF;
    const long long* p1 = positions + (size_t)b1 * N_HALF;
    for (int l = tid; l < N_HALF; l += 512) {
      const v2f y = sZ[l];
      const float m0 = (p0[l] != -1LL) ? 1.f : 0.f;
      const float m1 = (p1[l] != -1LL) ? 1.f : 0.f;
      o0[l] = (y.x + x0[l] * bv) * m0;
      o1[l] = (y.y + x1[l] * bv) * m1;
    }
    __syncthreads();   // sZ is re-filled next iteration
  }
}

extern "C" void kernel_launch(void* const* d_in, const int* in_sizes, int n_in,
                              void* d_out, int out_size, void* d_ws,
                              size_t ws_size, hipStream_t stream) {
  (void)in_sizes; (void)n_in; (void)out_size; (void)d_ws; (void)ws_size;
  const float*     inputs    = (const float*)d_in[0];
  const float*     filters   = (const float*)d_in[1];
  const float*     bias      = (const float*)d_in[2];
  const long long* positions = (const long long*)d_in[3];
  float*           out       = (float*)d_out;

  fft_long_conv_kernel<<<dim3(8 * 64), dim3(512), 0, stream>>>(
      inputs, filters, bias, positions, out);
}